// SequenceBeamSearch_38233798869780
// MI455X (gfx1250) — compile-verified
//
#include <hip/hip_runtime.h>
#include <math.h>

#define BB 16
#define KK 4
#define VV 32000
#define DD 512
#define LL 32
#define EOS_ID 2
#define NEG_INF_F 10000000.0f
#define ALPHA_F 0.6f
#define ROWS (BB*KK)     // 64
#define SEQ_LEN (LL+1)   // 33
#define NCHUNK (VV/128)  // 250 vocab chunks of 128 columns

typedef __attribute__((ext_vector_type(2))) float v2f;
typedef __attribute__((ext_vector_type(8))) float v8f;

// ---------------- init state ----------------
__global__ void k_init(const int* __restrict__ initial_ids, const float* __restrict__ init_h,
                       int* __restrict__ aliveSeq, float* __restrict__ aliveLp, float* __restrict__ cache,
                       int* __restrict__ finSeq, float* __restrict__ finSc, int* __restrict__ finFl) {
    int bk = blockIdx.x; int b = bk / KK; int k = bk % KK;
    int tid = threadIdx.x;
    for (int p = tid; p < SEQ_LEN; p += blockDim.x) {
        aliveSeq[bk*SEQ_LEN + p] = (p == 0) ? initial_ids[b] : 0;
        finSeq[bk*SEQ_LEN + p] = 0;
    }
    for (int d = tid; d < DD; d += blockDim.x)
        cache[bk*DD + d] = init_h[b*DD + d];
    if (tid == 0) {
        aliveLp[bk] = (k == 0) ? 0.0f : -NEG_INF_F;
        finSc[bk] = -NEG_INF_F;
        finFl[bk] = 0;
    }
}

// ---------------- h = tanh(cache + embed[last]) ----------------
__global__ void k_prep(const int* __restrict__ aliveSeq, const float* __restrict__ cache,
                       const float* __restrict__ embed, float* __restrict__ h, int step) {
    int bk = blockIdx.x; int tid = threadIdx.x;
    int last = aliveSeq[bk*SEQ_LEN + step];
    const float* e = embed + (size_t)last * DD;
    for (int d = tid; d < DD; d += blockDim.x)
        h[bk*DD + d] = tanhf(cache[bk*DD + d] + e[d]);
}

// ---------------- fused: logits tile via V_WMMA_F32_16X16X4_F32 + softmax/top-8 partials ----------------
// grid: (NCHUNK, 2), block 256 (8 waves). Block: 32 rows x 128 cols; no logits array.
// Emits per-(row,chunk): running max, sum-exp(rel max), top-8 (value, vocab id).
__global__ void __launch_bounds__(256) k_gemm(const float* __restrict__ h,
                                              const float* __restrict__ proj,
                                              float* __restrict__ pm, float* __restrict__ ps,
                                              float* __restrict__ pv, int* __restrict__ pi) {
    __shared__ float lh[32 * DD];                 // 64 KB: phase1 A panel, phase2 score/reduce scratch
    int tid = threadIdx.x;
    int mBase = blockIdx.y * 32;
    {
        const float4* src = (const float4*)(h + (size_t)mBase * DD);
        float4* dst = (float4*)lh;
        for (int idx = tid; idx < (32*DD)/4; idx += 256) dst[idx] = src[idx];
    }
    __syncthreads();

    int wave = tid >> 5;
    int lane = tid & 31;
    int half = lane >> 4;        // K/lane-half split for A and B operands
    int nIn  = lane & 15;        // N (and A-row) index within tile
    int nBase = blockIdx.x * 128 + wave * 16;

    v8f acc0 = {}; v8f acc1 = {};
    const float* pb = proj + (size_t)(2*half) * VV + nBase + nIn;

    for (int k0 = 0; k0 < DD; k0 += 4) {
        v2f bvec;
        bvec.x = pb[0];
        bvec.y = pb[VV];
        pb += (size_t)4 * VV;
        int aoff = k0 + 2*half;
        v2f a0 = *(const v2f*)&lh[( 0 + nIn)*DD + aoff];
        v2f a1 = *(const v2f*)&lh[(16 + nIn)*DD + aoff];
        acc0 = __builtin_amdgcn_wmma_f32_16x16x4_f32(false, a0, false, bvec, (short)0, acc0, false, false);
        acc1 = __builtin_amdgcn_wmma_f32_16x16x4_f32(false, a1, false, bvec, (short)0, acc1, false, false);
    }

    // ---- phase 2: stage tile scores in LDS (A panel no longer needed) ----
    __syncthreads();
    float* sc = lh;                       // [32][128] scores
    float* rm  = lh + 4096;               // [256] partial max
    float* rs  = lh + 4352;               // [256] partial sumexp
    float* rtv = lh + 4608;               // [256][8] partial top vals
    int*   rti = (int*)(lh + 6656);       // [256][8] partial top ids

    #pragma unroll
    for (int r = 0; r < 8; ++r) {
        sc[( 0 + r + 8*half)*128 + wave*16 + nIn] = acc0[r];
        sc[(16 + r + 8*half)*128 + wave*16 + nIn] = acc1[r];
    }
    __syncthreads();

    int rrow = tid >> 3;                  // 0..31 local row
    int seg  = tid & 7;                   // 8 threads per row, 16 cols each
    float lm = -INFINITY, ls = 0.0f;
    float tv[8]; int ti[8];
    #pragma unroll
    for (int j = 0; j < 8; ++j) { tv[j] = -INFINITY; ti[j] = 0; }
    int cBase = seg * 16;
    #pragma unroll
    for (int c = 0; c < 16; ++c) {
        float x = sc[rrow*128 + cBase + c];
        if (x > lm) { ls = ls * __expf(lm - x) + 1.0f; lm = x; }
        else        { ls += __expf(x - lm); }
        if (x > tv[7]) {
            int p = 7;
            while (p > 0 && x > tv[p-1]) { tv[p] = tv[p-1]; ti[p] = ti[p-1]; --p; }
            tv[p] = x; ti[p] = blockIdx.x * 128 + cBase + c;
        }
    }
    rm[tid] = lm; rs[tid] = ls;
    #pragma unroll
    for (int j = 0; j < 8; ++j) { rtv[tid*8+j] = tv[j]; rti[tid*8+j] = ti[j]; }
    __syncthreads();
    for (int s = 4; s > 0; s >>= 1) {
        if (seg < s) {
            int o = tid + s;
            float m1 = rm[tid], s1 = rs[tid];
            float m2 = rm[o],   s2 = rs[o];
            float M = fmaxf(m1, m2);
            float S = s1 * __expf(m1 - M) + s2 * __expf(m2 - M);
            rm[tid] = M; rs[tid] = S;
            float mv[8]; int mi[8];
            int p1 = 0, p2 = 0;
            #pragma unroll
            for (int j = 0; j < 8; ++j) {
                if (rtv[tid*8+p1] >= rtv[o*8+p2]) { mv[j] = rtv[tid*8+p1]; mi[j] = rti[tid*8+p1]; ++p1; }
                else                              { mv[j] = rtv[o*8+p2];   mi[j] = rti[o*8+p2];   ++p2; }
            }
            #pragma unroll
            for (int j = 0; j < 8; ++j) { rtv[tid*8+j] = mv[j]; rti[tid*8+j] = mi[j]; }
        }
        __syncthreads();
    }
    if (seg == 0) {
        int grow = mBase + rrow;
        size_t pidx = (size_t)grow * NCHUNK + blockIdx.x;
        pm[pidx] = rm[tid];
        ps[pidx] = rs[tid];
        #pragma unroll
        for (int j = 0; j < 8; ++j) { pv[pidx*8+j] = rtv[tid*8+j]; pi[pidx*8+j] = rti[tid*8+j]; }
    }
}

// ---------------- merge chunk partials per row: logZ + row top-8 ----------------
__global__ void k_merge(const float* __restrict__ pm, const float* __restrict__ ps,
                        const float* __restrict__ pv, const int* __restrict__ pi,
                        float* __restrict__ rowZ, float* __restrict__ rowV, int* __restrict__ rowI) {
    int row = blockIdx.x; int tid = threadIdx.x;
    float lm = -INFINITY, ls = 0.0f;
    float tv[8]; int ti[8];
    #pragma unroll
    for (int j = 0; j < 8; ++j) { tv[j] = -INFINITY; ti[j] = 0; }
    if (tid < NCHUNK) {
        size_t pidx = (size_t)row * NCHUNK + tid;
        lm = pm[pidx]; ls = ps[pidx];
        #pragma unroll
        for (int j = 0; j < 8; ++j) { tv[j] = pv[pidx*8+j]; ti[j] = pi[pidx*8+j]; }
    }
    __shared__ float sm[256], ss[256];
    __shared__ float stv[256][8];
    __shared__ int   sti[256][8];
    sm[tid] = lm; ss[tid] = ls;
    #pragma unroll
    for (int j = 0; j < 8; ++j) { stv[tid][j] = tv[j]; sti[tid][j] = ti[j]; }
    __syncthreads();
    for (int s = 128; s > 0; s >>= 1) {
        if (tid < s) {
            float m1 = sm[tid], s1 = ss[tid];
            float m2 = sm[tid+s], s2 = ss[tid+s];
            float M = fmaxf(m1, m2);
            float S = ((m1 > -INFINITY) ? s1 * __expf(m1 - M) : 0.0f)
                    + ((m2 > -INFINITY) ? s2 * __expf(m2 - M) : 0.0f);
            sm[tid] = M; ss[tid] = S;
            float mv[8]; int mi[8];
            int p1 = 0, p2 = 0;
            #pragma unroll
            for (int j = 0; j < 8; ++j) {
                if (stv[tid][p1] >= stv[tid+s][p2]) { mv[j] = stv[tid][p1]; mi[j] = sti[tid][p1]; ++p1; }
                else                                { mv[j] = stv[tid+s][p2]; mi[j] = sti[tid+s][p2]; ++p2; }
            }
            #pragma unroll
            for (int j = 0; j < 8; ++j) { stv[tid][j] = mv[j]; sti[tid][j] = mi[j]; }
        }
        __syncthreads();
    }
    if (tid == 0) {
        rowZ[row] = sm[0] + __logf(ss[0]);
        for (int j = 0; j < 8; ++j) { rowV[row*8+j] = stv[0][j]; rowI[row*8+j] = sti[0][j]; }
    }
}

// ---------------- per-batch selection + state update ----------------
__global__ void k_select(int step, const float* __restrict__ h,
                         const float* __restrict__ rowZ, const float* __restrict__ rowV, const int* __restrict__ rowI,
                         const int* __restrict__ aliveSeqIn, const float* __restrict__ aliveLpIn,
                         const int* __restrict__ finSeqIn, const float* __restrict__ finScIn, const int* __restrict__ finFlIn,
                         int* __restrict__ aliveSeqOut, float* __restrict__ aliveLpOut, float* __restrict__ cacheOut,
                         int* __restrict__ finSeqOut, float* __restrict__ finScOut, int* __restrict__ finFlOut) {
    int b = blockIdx.x; int tid = threadIdx.x;
    __shared__ float selLp[8]; __shared__ int selBeam[8]; __shared__ int selId[8]; __shared__ int selFl[8];
    __shared__ int aliveSrc[4]; __shared__ float lpOut[4];
    __shared__ int finSrc[4]; __shared__ float finScO[4]; __shared__ int finFlO[4];

    if (tid == 0) {
        float cl[32]; int ci[32];
        for (int k = 0; k < KK; ++k) {
            float adj = -rowZ[b*KK+k] + aliveLpIn[b*KK+k];
            for (int j = 0; j < 8; ++j) {
                cl[k*8+j] = rowV[(b*KK+k)*8+j] + adj;
                ci[k*8+j] = rowI[(b*KK+k)*8+j];
            }
        }
        bool used[32]; for (int c = 0; c < 32; ++c) used[c] = false;
        for (int s = 0; s < 8; ++s) {           // global top-8 over (k,v)
            float best = -INFINITY; int bi = 0;
            for (int c = 0; c < 32; ++c) if (!used[c] && cl[c] > best) { best = cl[c]; bi = c; }
            used[bi] = true;
            selLp[s] = best; selBeam[s] = bi >> 3; selId[s] = ci[bi];
            selFl[s] = (ci[bi] == EOS_ID) ? 1 : 0;
        }
        float asc[8]; bool u2[8];               // alive top-4 (EOS penalized)
        for (int j = 0; j < 8; ++j) { asc[j] = selLp[j] + (selFl[j] ? -NEG_INF_F : 0.0f); u2[j] = false; }
        for (int k = 0; k < KK; ++k) {
            float best = -INFINITY; int bj = 0;
            for (int j = 0; j < 8; ++j) if (!u2[j] && asc[j] > best) { best = asc[j]; bj = j; }
            u2[bj] = true; aliveSrc[k] = bj; lpOut[k] = best;
        }
        float ln = powf((5.0f + (float)(step + 1)) / 6.0f, ALPHA_F);
        float comb[12];                          // finished merge: 4 old + 8 new
        for (int k = 0; k < KK; ++k) comb[k] = finScIn[b*KK+k];
        for (int j = 0; j < 8; ++j) comb[4+j] = selLp[j] / ln + (selFl[j] ? 0.0f : -NEG_INF_F);
        bool u3[12]; for (int c = 0; c < 12; ++c) u3[c] = false;
        for (int k = 0; k < KK; ++k) {
            float best = -INFINITY; int bc = 0;
            for (int c = 0; c < 12; ++c) if (!u3[c] && comb[c] > best) { best = comb[c]; bc = c; }
            u3[bc] = true; finSrc[k] = bc; finScO[k] = best;
            finFlO[k] = (bc < 4) ? finFlIn[b*KK+bc] : selFl[bc-4];
        }
    }
    __syncthreads();

    for (int idx = tid; idx < KK*SEQ_LEN; idx += blockDim.x) {
        int k = idx / SEQ_LEN, p = idx % SEQ_LEN;
        int j = aliveSrc[k]; int beam = selBeam[j];
        aliveSeqOut[(b*KK+k)*SEQ_LEN + p] =
            (p == step + 1) ? selId[j] : aliveSeqIn[(b*KK+beam)*SEQ_LEN + p];
        int src = finSrc[k]; int outv;
        if (src < 4) outv = finSeqIn[(b*KK+src)*SEQ_LEN + p];
        else {
            int jj = src - 4;
            outv = (p == step + 1) ? selId[jj] : aliveSeqIn[(b*KK+selBeam[jj])*SEQ_LEN + p];
        }
        finSeqOut[(b*KK+k)*SEQ_LEN + p] = outv;
    }
    for (int idx = tid; idx < KK*DD; idx += blockDim.x) {
        int k = idx / DD, d = idx % DD;
        int beam = selBeam[aliveSrc[k]];
        cacheOut[(b*KK+k)*DD + d] = h[(b*KK+beam)*DD + d];
    }
    if (tid < KK) {
        aliveLpOut[b*KK+tid] = lpOut[tid];
        finScOut[b*KK+tid]  = finScO[tid];
        finFlOut[b*KK+tid]  = finFlO[tid];
    }
}

// ---------------- final output selection ----------------
__global__ void k_final(const int* __restrict__ aliveSeq, const float* __restrict__ aliveLp,
                        const int* __restrict__ finSeq, const float* __restrict__ finSc, const int* __restrict__ finFl,
                        float* __restrict__ out) {
    int b = blockIdx.x; int tid = threadIdx.x;
    int any = finFl[b*KK+0] | finFl[b*KK+1] | finFl[b*KK+2] | finFl[b*KK+3];
    for (int idx = tid; idx < KK*SEQ_LEN; idx += blockDim.x) {
        int g = b*KK*SEQ_LEN + idx;
        out[g] = (float)(any ? finSeq[g] : aliveSeq[g]);
    }
    if (tid < KK)
        out[BB*KK*SEQ_LEN + b*KK + tid] = any ? finSc[b*KK+tid] : aliveLp[b*KK+tid];
}

extern "C" void kernel_launch(void* const* d_in, const int* in_sizes, int n_in,
                              void* d_out, int out_size, void* d_ws, size_t ws_size,
                              hipStream_t stream) {
    const int*   initial_ids = (const int*)d_in[0];
    const float* init_h      = (const float*)d_in[1];
    const float* embed       = (const float*)d_in[2];
    const float* proj        = (const float*)d_in[3];
    float* out = (float*)d_out;
    (void)in_sizes; (void)n_in; (void)out_size; (void)ws_size;

    char* ws = (char*)d_ws;
    size_t off = 0;
    auto alloc = [&](size_t bytes) -> char* {
        char* p = ws + off;
        off = (off + bytes + 255) & ~(size_t)255;
        return p;
    };
    float* h    = (float*)alloc((size_t)ROWS * DD * 4);
    float* pm   = (float*)alloc((size_t)ROWS * NCHUNK * 4);
    float* ps   = (float*)alloc((size_t)ROWS * NCHUNK * 4);
    float* pv   = (float*)alloc((size_t)ROWS * NCHUNK * 8 * 4);
    int*   pi   = (int*)alloc((size_t)ROWS * NCHUNK * 8 * 4);
    float* rowZ = (float*)alloc(ROWS * 4);
    float* rowV = (float*)alloc(ROWS * 8 * 4);
    int*   rowI = (int*)alloc(ROWS * 8 * 4);
    int* aliveSeq[2]; float* aliveLp[2]; float* cache[2];
    int* finSeq[2];   float* finSc[2];   int* finFl[2];
    for (int s = 0; s < 2; ++s) {
        aliveSeq[s] = (int*)alloc(ROWS * SEQ_LEN * 4);
        aliveLp[s]  = (float*)alloc(ROWS * 4);
        cache[s]    = (float*)alloc((size_t)ROWS * DD * 4);
        finSeq[s]   = (int*)alloc(ROWS * SEQ_LEN * 4);
        finSc[s]    = (float*)alloc(ROWS * 4);
        finFl[s]    = (int*)alloc(ROWS * 4);
    }

    k_init<<<ROWS, 128, 0, stream>>>(initial_ids, init_h,
                                     aliveSeq[0], aliveLp[0], cache[0],
                                     finSeq[0], finSc[0], finFl[0]);
    int cur = 0;
    for (int i = 0; i < LL; ++i) {
        int nxt = cur ^ 1;
        k_prep<<<ROWS, 256, 0, stream>>>(aliveSeq[cur], cache[cur], embed, h, i);
        k_gemm<<<dim3(NCHUNK, 2), 256, 0, stream>>>(h, proj, pm, ps, pv, pi);
        k_merge<<<ROWS, 256, 0, stream>>>(pm, ps, pv, pi, rowZ, rowV, rowI);
        k_select<<<BB, 128, 0, stream>>>(i, h, rowZ, rowV, rowI,
                                         aliveSeq[cur], aliveLp[cur],
                                         finSeq[cur], finSc[cur], finFl[cur],
                                         aliveSeq[nxt], aliveLp[nxt], cache[nxt],
                                         finSeq[nxt], finSc[nxt], finFl[nxt]);
        cur = nxt;
    }
    k_final<<<BB, 128, 0, stream>>>(aliveSeq[cur], aliveLp[cur],
                                    finSeq[cur], finSc[cur], finFl[cur], out);
}